// DensenetLossCompute_50620484551284
// MI455X (gfx1250) — compile-verified
//
#include <hip/hip_runtime.h>

typedef float v2f __attribute__((ext_vector_type(2)));
typedef float v8f __attribute__((ext_vector_type(8)));

#define LN2F 0.69314718055994530942f

// Row-wise weighted BCE:  out[r] = -ln2 * sum_c(w_c * log2(sel_c + eps)) / sum_c(w_c)
// with w_c = wneg_c * (1 + y_c), wneg = [1,1,1,1,1,1,1,7], sel = y ? p : 1-p.
//
// One wave32 handles a 16-row tile. Lanes 0-15 load cols {0,1,4,5} of rows
// 0..15, lanes 16-31 load cols {2,3,6,7} -- this is exactly the B-matrix
// (4x16, f32) VGPR layout for V_WMMA_F32_16X16X4_F32 (V0/V1 = K0,K1 in lanes
// 0-15 and K2,K3 in lanes 16-31). With A == ones(16x4), D[m][n] = sum_k B[k][n],
// so one chained WMMA pair produces all 16 row-sums; a second pair sums the
// weights. D VGPR0 holds row-sum(row = base + (lane&15)) in every lane.
__global__ __launch_bounds__(256) void wbce_wmma_kernel(
    const float* __restrict__ logits,   // [rows][8]
    const float* __restrict__ targets,  // [rows][8]
    float* __restrict__ out,            // [rows]
    int rows) {
  const int lane = threadIdx.x & 31;
  const int wave = threadIdx.x >> 5;
  const int r    = lane & 15;
  const int half = lane >> 4;

  const int base = (blockIdx.x * 8 + wave) * 16;
  const int row  = base + r;
  // Clamp so EXEC stays all-ones through the WMMAs (tail rows recompute a
  // valid row; only the store is predicated).
  const int lrow = (row < rows) ? row : (rows - 1);

  const int c0 = 2 * half;  // this lane's first column (K position 2*half)
  const float* pL = logits  + (size_t)lrow * 8;
  const float* pT = targets + (size_t)lrow * 8;

  // 4x global_load_b64: cols {c0,c0+1} and {c0+4,c0+5} of each input.
  const v2f pA = *(const v2f*)(pL + c0);
  const v2f pB = *(const v2f*)(pL + c0 + 4);
  const v2f tA = *(const v2f*)(pT + c0);
  const v2f tB = *(const v2f*)(pT + c0 + 4);

  // Column 7 (wneg = 7) is slot 3 (c0+5) of the half==1 lanes.
  const float wn3 = half ? 7.0f : 1.0f;

  // Per-element: q = y ? p : 1-p ; lg = log2(q + eps) ; w = wneg*(1+y)
  float v[4], w[4];
  {
    const float p[4] = {pA[0], pA[1], pB[0], pB[1]};
    const float y[4] = {tA[0], tA[1], tB[0], tB[1]};
    const float wn[4] = {1.0f, 1.0f, 1.0f, wn3};
#pragma unroll
    for (int i = 0; i < 4; ++i) {
      const float q  = (y[i] > 0.5f) ? p[i] : (1.0f - p[i]);
      const float lg = __builtin_amdgcn_logf(q + 1e-8f);  // v_log_f32 (log2)
      w[i] = wn[i] * (1.0f + y[i]);
      v[i] = w[i] * lg;
    }
  }

  // WMMA B operands: {slot0, slot1} covers cols 0-3 across the wave,
  // {slot2, slot3} covers cols 4-7.
  const v2f a   = {1.0f, 1.0f};
  const v2f bL1 = {v[0], v[1]};
  const v2f bL2 = {v[2], v[3]};
  const v2f bW1 = {w[0], w[1]};
  const v2f bW2 = {w[2], w[3]};

  v8f accL = {};
  accL = __builtin_amdgcn_wmma_f32_16x16x4_f32(false, a, false, bL1,
                                               (short)0, accL, false, false);
  accL = __builtin_amdgcn_wmma_f32_16x16x4_f32(false, a, false, bL2,
                                               (short)0, accL, false, false);
  v8f accW = {};
  accW = __builtin_amdgcn_wmma_f32_16x16x4_f32(false, a, false, bW1,
                                               (short)0, accW, false, false);
  accW = __builtin_amdgcn_wmma_f32_16x16x4_f32(false, a, false, bW2,
                                               (short)0, accW, false, false);

  const float sumL = accL[0];  // row-sum for row base+(lane&15), all lanes
  const float sumW = accW[0];

  if (half == 0 && row < rows) {
    out[row] = (-LN2F) * sumL / sumW;
  }
}

extern "C" void kernel_launch(void* const* d_in, const int* in_sizes, int n_in,
                              void* d_out, int out_size, void* d_ws, size_t ws_size,
                              hipStream_t stream) {
  const float* logits  = (const float*)d_in[0];
  const float* targets = (const float*)d_in[1];
  float* out = (float*)d_out;

  const int rows = in_sizes[0] / 8;           // N (C == 8)
  const int blocks = (rows + 127) / 128;      // 8 waves/block * 16 rows/wave

  wbce_wmma_kernel<<<blocks, 256, 0, stream>>>(logits, targets, out, rows);
}